// CustomMaskedConv1D_5033701671547
// MI455X (gfx1250) — compile-verified
//
#include <hip/hip_runtime.h>

typedef float v2f __attribute__((ext_vector_type(2)));
typedef float v8f __attribute__((ext_vector_type(8)));

#define BATCH   8
#define SEQ     2048
#define CIN     512
#define GROUPS  8
#define ICPG    64
#define OCPG    64
#define KTAPS   5
#define SCHUNK  128              // seq rows per block
#define XROWS   (SCHUNK + 4)     // 132 staged rows (halo of 2 each side)
#define XPITCH  68               // padded X row pitch (floats): 68 % 64 == 4 -> conflict-free b64 reads
#define KFLAT   (ICPG * KTAPS)   // 320
#define WPITCH  324              // padded W row pitch (floats): 324 % 64 == 4

// out[b, s, g, o] = sum_{t, i} mask(b,s,t) * in[b, s+t-2, g*64+i] * kern[g, o, i, t]
// Per group: GEMM (S*B x 320) @ (320 x 64), computed as 16x16x4 f32 WMMA tiles.
__global__ __launch_bounds__(256)
void masked_gconv1d_wmma(const float* __restrict__ in,
                         const int*   __restrict__ pos,
                         const float* __restrict__ kern,
                         float*       __restrict__ out) {
  extern __shared__ char smem[];
  float* Wl = (float*)smem;                 // [OCPG][WPITCH]   (kfl = tap*64 + ic)
  float* Xl = Wl + OCPG * WPITCH;           // [XROWS][XPITCH]
  int*   Pl = (int*)(Xl + XROWS * XPITCH);  // [XROWS]

  const int tid = threadIdx.x;
  const int s0  = blockIdx.x * SCHUNK;
  const int g   = blockIdx.y;
  const int b   = blockIdx.z;

  // ---- stage weights: global (g,o,i,t) contiguous -> LDS [o][t*64+i] (coalesced reads)
  const float* Wg = kern + (size_t)g * OCPG * KFLAT;
  for (int e = tid; e < OCPG * KFLAT; e += 256) {
    int o  = e / KFLAT;
    int r  = e - o * KFLAT;        // = i*5 + t  (global contiguous within an o-row)
    int i  = r / KTAPS;
    int t  = r - i * KTAPS;
    Wl[o * WPITCH + t * ICPG + i] = Wg[e];
  }

  // ---- stage input rows [s0-2, s0+129], zero-padded out of range
  const size_t inb = (size_t)b * SEQ * CIN + (size_t)g * ICPG;
  for (int e = tid; e < XROWS * ICPG; e += 256) {
    int r = e >> 6;
    int c = e & 63;
    int s = s0 + r - 2;
    float v = 0.0f;
    if (s >= 0 && s < SEQ) v = in[inb + (size_t)s * CIN + c];
    Xl[r * XPITCH + c] = v;
  }

  // ---- stage positions (sentinel in halo; halo inputs are zero so mask value is moot there)
  for (int r = tid; r < XROWS; r += 256) {
    int s = s0 + r - 2;
    Pl[r] = (s >= 0 && s < SEQ) ? pos[(size_t)b * SEQ + s] : (int)0x80000000;
  }
  __syncthreads();

  const int wave = tid >> 5;
  const int lane = tid & 31;
  const int row  = lane & 15;    // output seq row within the 16-row tile / B-matrix column
  const int hi   = lane >> 4;    // half-wave select
  const int klo  = hi * 2;       // A/B fragments: lanes 0-15 hold K={0,1}, lanes 16-31 K={2,3}

  // per-lane tap masks for this output row (positions must be contiguous across the window)
  float mv[KTAPS];
  const int rc = wave * 16 + row + 2;
  const int pc = Pl[rc];
#pragma unroll
  for (int t = 0; t < KTAPS; ++t)
    mv[t] = (Pl[rc + t - 2] == pc + (t - 2)) ? 1.0f : 0.0f;

  v8f acc[4] = {};   // four 16x16 f32 accumulators covering oc = 0..63

#pragma unroll
  for (int t = 0; t < KTAPS; ++t) {
    const float m = mv[t];
    const float* xr = &Xl[(wave * 16 + row + t) * XPITCH + klo];
    const float* wr = &Wl[row * WPITCH + t * ICPG + klo];
#pragma unroll 4
    for (int kb = 0; kb < 16; ++kb) {       // 16 K-blocks of 4 per tap
      v2f a = *(const v2f*)(xr + kb * 4);   // A: 16x4 f32 fragment
      a.x *= m;
      a.y *= m;
#pragma unroll
      for (int n = 0; n < 4; ++n) {         // 4 N-tiles share the A fragment
        v2f bv = *(const v2f*)(wr + n * 16 * WPITCH + kb * 4);  // B: 4x16 f32 fragment
        acc[n] = __builtin_amdgcn_wmma_f32_16x16x4_f32(
            /*neg_a=*/false, a, /*neg_b=*/false, bv,
            /*c_mod=*/(short)0, acc[n], /*reuse_a=*/false, /*reuse_b=*/false);
      }
    }
  }

  // ---- store: C/D layout is VGPR v -> M = v + hi*8, N = lane&15
  const size_t ob = ((size_t)b * SEQ + (size_t)(s0 + wave * 16)) * CIN
                  + (size_t)g * OCPG + row;
#pragma unroll
  for (int n = 0; n < 4; ++n) {
#pragma unroll
    for (int v = 0; v < 8; ++v) {
      int M = v + hi * 8;
      out[ob + (size_t)M * CIN + n * 16] = acc[n][v];
    }
  }
}

extern "C" void kernel_launch(void* const* d_in, const int* in_sizes, int n_in,
                              void* d_out, int out_size, void* d_ws, size_t ws_size,
                              hipStream_t stream) {
  (void)in_sizes; (void)n_in; (void)out_size; (void)d_ws; (void)ws_size;
  const float* in   = (const float*)d_in[0];   // (B, S, 512) f32
  const int*   pos  = (const int*)d_in[1];     // (B, S) i32
  const float* kern = (const float*)d_in[2];   // (G, 64, 64, 5) f32
  float*       out  = (float*)d_out;           // (B, S, G, 64) f32

  dim3 grid(SEQ / SCHUNK, GROUPS, BATCH);      // 16 x 8 x 8 = 1024 blocks
  dim3 block(256);                             // 8 waves of 32
  size_t smem = (size_t)(OCPG * WPITCH + XROWS * XPITCH) * sizeof(float)
              + (size_t)XROWS * sizeof(int);   // ~117 KB of the 320 KB WGP LDS
  masked_gconv1d_wmma<<<grid, block, smem, stream>>>(in, pos, kern, out);
}